// BatchSoftmaxNomax_36060545417487
// MI455X (gfx1250) — compile-verified
//
#include <hip/hip_runtime.h>
#include <math.h>

typedef __attribute__((ext_vector_type(2))) float v2f;
typedef __attribute__((ext_vector_type(8))) float v8f;

#define BB 256      // batch
#define NSPAN 32    // spans per sample (both tensors)
#define DD 512      // feature dim

// ---------------------------------------------------------------------------
// Kernel 1: mean over axis 1 of (B, 32, D) -> (B, D).  Memory-bound streaming
// pass: thread t handles output element (b, d); consecutive threads touch
// consecutive d, so every global_load is a fully coalesced 512-float row.
// ---------------------------------------------------------------------------
__global__ void mean_spans_kernel(const float* __restrict__ in,
                                  float* __restrict__ out) {
    int t = blockIdx.x * blockDim.x + threadIdx.x;   // 0 .. B*D-1
    int b = t >> 9;                                   // / 512
    int d = t & (DD - 1);
    const float* p = in + (size_t)b * (NSPAN * DD) + d;
    float s = 0.0f;
#pragma unroll
    for (int n = 0; n < NSPAN; ++n) s += p[n * DD];
    out[t] = s * (1.0f / NSPAN);
}

// ---------------------------------------------------------------------------
// Kernel 2: logits = U @ V^T  (256x512 @ 512x256) in full f32 via
// V_WMMA_F32_16X16X4_F32.  One wave (32 lanes) per 16x16 output tile,
// 16x16 = 256 tiles, 512/4 = 128 WMMA issues per tile.
//
// A fragment (16x4 f32): lanes 0-15 -> M=lane, K={0,1}; lanes 16-31 ->
// M=lane-16, K={2,3}  => float2 load at U[row*D + k + (lane>=16 ? 2 : 0)].
// B fragment (4x16) mirrors it with V indexed by the output column
// (B[d][n] = V[n][d]).  C/D: VGPR j -> row j + 8*(lane>=16), col lane&15.
// ---------------------------------------------------------------------------
__global__ void gemm_wmma_f32_kernel(const float* __restrict__ U,
                                     const float* __restrict__ V,
                                     float* __restrict__ L) {
    const int tile = blockIdx.x;        // 0..255
    const int mt = tile >> 4;
    const int nt = tile & 15;
    const int lane = threadIdx.x;       // wave32: 0..31, EXEC all ones
    const int half = lane >> 4;         // 0 or 1 (selects K pair)
    const int l15  = lane & 15;

    const float* urow = U + (size_t)(mt * 16 + l15) * DD + half * 2;
    const float* vrow = V + (size_t)(nt * 16 + l15) * DD + half * 2;

    v8f c = {};
#pragma unroll 8
    for (int k = 0; k < DD; k += 4) {
        v2f a = *(const v2f*)(urow + k);
        v2f b = *(const v2f*)(vrow + k);
        c = __builtin_amdgcn_wmma_f32_16x16x4_f32(
                /*neg_a=*/false, a, /*neg_b=*/false, b,
                /*c_mod=*/(short)0, c, /*reuse_a=*/false, /*reuse_b=*/false);
    }

    const int col   = nt * 16 + l15;
    const int rbase = mt * 16 + half * 8;
#pragma unroll
    for (int j = 0; j < 8; ++j)
        L[(size_t)(rbase + j) * BB + col] = c[j];
}

// ---------------------------------------------------------------------------
// Kernel 3: loss = mean_b [ logsumexp(logits[b,:]) - logits[b,b] ].
// One block, 256 threads, one row per thread (256 KB, L2-resident epilogue).
// ---------------------------------------------------------------------------
__global__ void ce_loss_kernel(const float* __restrict__ L,
                               float* __restrict__ out) {
    __shared__ float red[BB];
    const int b = threadIdx.x;
    const float* row = L + (size_t)b * BB;

    float m = -INFINITY;
    for (int c = 0; c < BB; ++c) m = fmaxf(m, row[c]);
    float s = 0.0f;
    for (int c = 0; c < BB; ++c) s += __expf(row[c] - m);
    red[b] = (m + __logf(s)) - row[b];
    __syncthreads();

    for (int off = BB / 2; off > 0; off >>= 1) {
        if (b < off) red[b] += red[b + off];
        __syncthreads();
    }
    if (b == 0) out[0] = red[0] * (1.0f / BB);
}

// ---------------------------------------------------------------------------
// Launch: d_in[0] = face_j (B,32,D), d_in[1] = ner_j (B,32,D); output scalar.
// Workspace: U (512 KB) | V (512 KB) | logits (256 KB).
// ---------------------------------------------------------------------------
extern "C" void kernel_launch(void* const* d_in, const int* in_sizes, int n_in,
                              void* d_out, int out_size, void* d_ws, size_t ws_size,
                              hipStream_t stream) {
    const float* face = (const float*)d_in[0];
    const float* ner  = (const float*)d_in[1];

    float* U = (float*)d_ws;          // (256, 512) mean of ner spans
    float* V = U + BB * DD;           // (256, 512) mean of face spans
    float* L = V + BB * DD;           // (256, 256) logits

    const int nElem  = BB * DD;               // 131072
    const int blocks = nElem / 256;           // 512

    mean_spans_kernel<<<blocks, 256, 0, stream>>>(ner,  U);
    mean_spans_kernel<<<blocks, 256, 0, stream>>>(face, V);
    gemm_wmma_f32_kernel<<<BB * BB / (16 * 16), 32, 0, stream>>>(U, V, L);
    ce_loss_kernel<<<1, BB, 0, stream>>>(L, (float*)d_out);
}